// DynamicsNetwork_9259949490920
// MI455X (gfx1250) — compile-verified
//
#include <hip/hip_runtime.h>

typedef float v2f __attribute__((ext_vector_type(2)));
typedef float v8f __attribute__((ext_vector_type(8)));

#define OBS_DIM   64
#define ACT_DIM   16
#define IN_DIM    320
#define ENV_B     32
#define STEP_B    1024
#define NROWS     32          // batch rows per workgroup (2 WMMA row-tiles)
#define LDSTRIDE  1028        // 4-bank row offset -> conflict-free 16x4 A fetch (64 distinct banks)
#define NTHREADS  256
#define NWAVES    8
#define OUT_LIMIT 100.0f

// One layer: Y[32,N] = SiLU/clip( X[32,K] * W^T + b ).
// Each wave owns a 16-wide column tile and computes BOTH 16-row tiles in one
// K-pass so every weight float2 feeds two WMMAs (4 independent accum chains).
template<int K, int N, bool FINAL>
__device__ __forceinline__ void layer_gemm(const float* __restrict__ Wl,   // [N,K] row-major (global)
                                           const float* __restrict__ bl,   // [N]   (global)
                                           const float* __restrict__ bin,  // LDS  [32][LDSTRIDE]
                                           float*       __restrict__ bout, // LDS  [32][LDSTRIDE]
                                           float*       __restrict__ out,  // global (FINAL only)
                                           int row0, int env,
                                           int wave, int lr, int half) {
    constexpr int NCT = N / 16;
    // A operand (16x4 f32 layout): lane L holds row M=L&15, K = k0 + 2*(L>>4) .. +1
    const float* a0row = bin + (size_t)lr        * LDSTRIDE + 2 * half;   // row-tile 0 (M 0-15)
    const float* a1row = bin + (size_t)(16 + lr) * LDSTRIDE + 2 * half;   // row-tile 1 (M 16-31)

    for (int ct = wave; ct < NCT; ct += NWAVES) {
        const int oBase = ct * 16;
        // B operand: B[k][n] = W[oBase+n][k]; per-lane float2 from weight row
        const float* wrow = Wl + (size_t)(oBase + lr) * K + 2 * half;

        v8f accA0 = {0.f,0.f,0.f,0.f,0.f,0.f,0.f,0.f};
        v8f accA1 = {0.f,0.f,0.f,0.f,0.f,0.f,0.f,0.f};
        v8f accB0 = {0.f,0.f,0.f,0.f,0.f,0.f,0.f,0.f};
        v8f accB1 = {0.f,0.f,0.f,0.f,0.f,0.f,0.f,0.f};

        #pragma unroll 4
        for (int k = 0; k < K; k += 8) {
            v2f w0  = *(const v2f*)(wrow  + k);
            v2f w1  = *(const v2f*)(wrow  + k + 4);
            v2f aA0 = *(const v2f*)(a0row + k);
            v2f aA1 = *(const v2f*)(a0row + k + 4);
            v2f aB0 = *(const v2f*)(a1row + k);
            v2f aB1 = *(const v2f*)(a1row + k + 4);
            accA0 = __builtin_amdgcn_wmma_f32_16x16x4_f32(false, aA0, false, w0, (short)0, accA0, false, false);
            accB0 = __builtin_amdgcn_wmma_f32_16x16x4_f32(false, aB0, false, w0, (short)0, accB0, false, false);
            accA1 = __builtin_amdgcn_wmma_f32_16x16x4_f32(false, aA1, false, w1, (short)0, accA1, false, false);
            accB1 = __builtin_amdgcn_wmma_f32_16x16x4_f32(false, aB1, false, w1, (short)0, accB1, false, false);
        }

        const float bias = bl[oBase + lr];

        #pragma unroll
        for (int r = 0; r < 8; ++r) {
            float vA = accA0[r] + accA1[r] + bias;
            float vB = accB0[r] + accB1[r] + bias;
            const int m = r + half * 8;           // C layout: lanes>=16 carry M+8
            if constexpr (FINAL) {
                vA = fminf(fmaxf(vA, -OUT_LIMIT), OUT_LIMIT);
                vB = fminf(fmaxf(vB, -OUT_LIMIT), OUT_LIMIT);
                const int gA = (row0 + m)      * ENV_B + env;   // back to [step*32+env]
                const int gB = (row0 + 16 + m) * ENV_B + env;
                out[(size_t)gA * OBS_DIM + oBase + lr] = vA;
                out[(size_t)gB * OBS_DIM + oBase + lr] = vB;
            } else {
                vA = vA / (1.0f + __expf(-vA));   // SiLU
                vB = vB / (1.0f + __expf(-vB));
                bout[(size_t)m        * LDSTRIDE + oBase + lr] = vA;
                bout[(size_t)(16 + m) * LDSTRIDE + oBase + lr] = vB;
            }
        }
    }
}

__global__ __launch_bounds__(NTHREADS)
void dynnet_fused_kernel(const float* __restrict__ obs,
                         const float* __restrict__ act,
                         const float* __restrict__ W0, const float* __restrict__ b0,
                         const float* __restrict__ W1, const float* __restrict__ b1,
                         const float* __restrict__ W2, const float* __restrict__ b2,
                         const float* __restrict__ W3, const float* __restrict__ b3,
                         const int*   __restrict__ batch_idx,
                         float*       __restrict__ out) {
    extern __shared__ float smem[];
    float* bufA = smem;                       // activation ping
    float* bufB = smem + NROWS * LDSTRIDE;    // activation pong

    const int env    = blockIdx.x >> 5;       // same-env blocks adjacent -> L2 weight reuse
    const int rowblk = blockIdx.x & 31;
    const int row0   = rowblk * NROWS;
    const int tid    = threadIdx.x;
    const int lane   = tid & 31;              // wave32
    const int wave   = tid >> 5;
    const int lr     = lane & 15;
    const int half   = lane >> 4;

    const int env_start = (batch_idx[0] / (ENV_B * STEP_B)) * ENV_B;
    const size_t e = (size_t)(env_start + env);

    // ---- Stage layer-0 input (concat obs|action) into bufA, coalesced float4 ----
    for (int i = tid; i < NROWS * 80; i += NTHREADS) {
        const int r = i / 80;            // local row
        const int q = i % 80;            // float4 slot within the 320-wide row
        const int gRow = (row0 + r) * ENV_B + env;
        float4 v;
        if (q < 64) v = ((const float4*)(obs + (size_t)gRow * (OBS_DIM * 4)))[q];
        else        v = ((const float4*)(act + (size_t)gRow * (ACT_DIM * 4)))[q - 64];
        *(float4*)(bufA + (size_t)r * LDSTRIDE + q * 4) = v;
    }
    __syncthreads();

    // Layers unrolled by hand: each weight pointer traces directly to a kernarg
    // so address-space inference emits GLOBAL (not FLAT) loads.
    layer_gemm<IN_DIM, 1024, false>(W0 + e * 1024 * IN_DIM, b0 + e * 1024,
                                    bufA, bufB, nullptr, row0, env, wave, lr, half);
    __syncthreads();
    layer_gemm<1024, 1024, false>(W1 + e * 1024 * 1024, b1 + e * 1024,
                                  bufB, bufA, nullptr, row0, env, wave, lr, half);
    __syncthreads();
    layer_gemm<1024, 1024, false>(W2 + e * 1024 * 1024, b2 + e * 1024,
                                  bufA, bufB, nullptr, row0, env, wave, lr, half);
    __syncthreads();
    layer_gemm<1024, OBS_DIM, true>(W3 + e * OBS_DIM * 1024, b3 + e * OBS_DIM,
                                    bufB, nullptr, out, row0, env, wave, lr, half);
}

extern "C" void kernel_launch(void* const* d_in, const int* in_sizes, int n_in,
                              void* d_out, int out_size, void* d_ws, size_t ws_size,
                              hipStream_t stream) {
    (void)in_sizes; (void)n_in; (void)d_ws; (void)ws_size; (void)out_size;
    const float* obs = (const float*)d_in[0];
    const float* act = (const float*)d_in[1];
    const float* W0  = (const float*)d_in[2];
    const float* b0  = (const float*)d_in[3];
    const float* W1  = (const float*)d_in[4];
    const float* b1  = (const float*)d_in[5];
    const float* W2  = (const float*)d_in[6];
    const float* b2  = (const float*)d_in[7];
    const float* W3  = (const float*)d_in[8];
    const float* b3  = (const float*)d_in[9];
    const int*   bidx = (const int*)d_in[10];
    float* out = (float*)d_out;

    const size_t smem = (size_t)2 * NROWS * LDSTRIDE * sizeof(float);  // ~257 KB < 320 KB WGP LDS
    (void)hipFuncSetAttribute((const void*)dynnet_fused_kernel,
                              hipFuncAttributeMaxDynamicSharedMemorySize, (int)smem);

    dim3 grid(ENV_B * (STEP_B / NROWS));   // 32 envs x 32 row-blocks = 1024 WGs
    dim3 block(NTHREADS);
    dynnet_fused_kernel<<<grid, block, smem, stream>>>(
        obs, act, W0, b0, W1, b1, W2, b2, W3, b3, bidx, out);
}